// SSIM_5669356833977
// MI455X (gfx1250) — compile-verified
//
#include <hip/hip_runtime.h>

// CDNA5 / gfx1250 SSIM: 7x7 box sums as banded-ones matmuls on
// V_WMMA_F32_16X16X4_F32. One wave32 per 16x32 output tile.
//
//   Out(16x32) = V(16x24) * P(24x48) * Wh(48x32)   per map P in {x,y,xx,yy,xy}
//
// V / Wh are 0/1 band matrices (7-tap box): band(r,c) = 1 iff 0 <= r-c <= 6.
// Pad rows/cols carry zero band weight, so the halo fill needs no zero-masking:
// interior tiles load directly, edge tiles only clamp addresses (duplicated
// boundary values feed only epilogue-masked outputs).

typedef float v2f __attribute__((ext_vector_type(2)));
typedef float v8f __attribute__((ext_vector_type(8)));

#define H_IMG 640
#define W_IMG 640
#define OUT_H 634          // 640 - 7 + 1
#define OUT_W 634
#define NB    64
#define SXW   48           // sX/sY row stride (dwords)
#define T1W   50           // sT1 row stride (dwords): bank-conflict-free A reads
#define NCELL 2048         // partial-sum cells in d_ws
#define TX    20           // x tiles (32 wide)
#define TY    40           // y tiles (16 tall)

__global__ void ssim_zero_kernel(float* ws) {
    int i = blockIdx.x * blockDim.x + threadIdx.x;
    if (i < NCELL) ws[i] = 0.0f;
}

__global__ __launch_bounds__(256)
void ssim_final_kernel(const float* __restrict__ ws, float* __restrict__ out) {
    __shared__ float red[256];
    const int t = threadIdx.x;
    float s = 0.0f;
    #pragma unroll
    for (int j = 0; j < NCELL / 256; ++j) s += ws[t + 256 * j];
    red[t] = s;
    __syncthreads();
    #pragma unroll
    for (int off = 128; off > 0; off >>= 1) {
        if (t < off) red[t] += red[t + off];
        __syncthreads();
    }
    if (t == 0) out[0] = red[0] / (float)((double)NB * (double)OUT_H * (double)OUT_W);
}

template<bool EDGE>
__device__ __forceinline__
void ssim_tile_body(const float* __restrict__ X, const float* __restrict__ Y,
                    const float* __restrict__ DR, float* __restrict__ ws,
                    int tx, int ty, int b)
{
    __shared__ float sX[24 * SXW];
    __shared__ float sY[24 * SXW];
    __shared__ float sT1[16 * T1W];

    const int lane = threadIdx.x;       // 0..31 (wave32)
    const int hi   = lane >> 4;
    const int lm   = lane & 15;
    const int r0   = ty * 16;
    const int c0   = tx * 32;
    const size_t base = (size_t)b * H_IMG * W_IMG;

    // ---- Per-lane columns (clamped only on edge tiles) ----
    const int col1 = c0 + lane;           // tile cols 0..31
    const int col2 = c0 + 32 + lm;        // tile cols 32..47 (upper lanes duplicate: benign)
    int cc1 = col1, cc2 = col2;
    if constexpr (EDGE) {
        cc1 = col1 < (W_IMG - 1) ? col1 : (W_IMG - 1);
        cc2 = col2 < (W_IMG - 1) ? col2 : (W_IMG - 1);
    }

    // ---- Fill 22 halo rows; zero rows 22..23 (zero band weight, just needs finite) ----
    #pragma unroll
    for (int i = 0; i < 22; ++i) {
        int gr = r0 + i;
        if constexpr (EDGE) gr = gr < (H_IMG - 1) ? gr : (H_IMG - 1);
        const size_t rb = base + (size_t)gr * W_IMG;
        sX[i * SXW + lane]    = X[rb + cc1];
        sY[i * SXW + lane]    = Y[rb + cc1];
        sX[i * SXW + 32 + lm] = X[rb + cc2];
        sY[i * SXW + 32 + lm] = Y[rb + cc2];
    }
    #pragma unroll
    for (int i = 22; i < 24; ++i) {
        sX[i * SXW + lane] = 0.0f;  sX[i * SXW + 32 + lm] = 0.0f;
        sY[i * SXW + lane] = 0.0f;  sY[i * SXW + 32 + lm] = 0.0f;
    }
    __syncthreads();

    // ---- Hoisted band fragments: band(r,c) = 1 iff 0 <= r-c <= 6 ----
    // Serve as stage-1 A (A[m][k] = band(k, m)) and stage-2 B (B[k][j] = band(k, j)).
    v2f bnd[6];
    #pragma unroll
    for (int kk = 0; kk < 6; ++kk) {
        const int k0 = 4 * kk + 2 * hi;
        bnd[kk].x = ((unsigned)(k0     - lm) <= 6u) ? 1.0f : 0.0f;
        bnd[kk].y = ((unsigned)(k0 + 1 - lm) <= 6u) ? 1.0f : 0.0f;
    }

    // ---- Stage 1: T1[map] = V(16x24) * P(24x48), three 16-wide column groups ----
    v8f acc1[5][3];
    #pragma unroll
    for (int m = 0; m < 5; ++m)
        #pragma unroll
        for (int g = 0; g < 3; ++g)
            acc1[m][g] = {};

    #pragma unroll
    for (int kk = 0; kk < 6; ++kk) {
        const int rk = 4 * kk + 2 * hi;
        #pragma unroll
        for (int g = 0; g < 3; ++g) {
            const int c = 16 * g + lm;
            const float x0 = sX[rk * SXW + c],       y0 = sY[rk * SXW + c];
            const float x1 = sX[(rk + 1) * SXW + c], y1 = sY[(rk + 1) * SXW + c];
            v2f bx, by, bxx, byy, bxy;
            bx.x  = x0;      bx.y  = x1;
            by.x  = y0;      by.y  = y1;
            bxx.x = x0 * x0; bxx.y = x1 * x1;
            byy.x = y0 * y0; byy.y = y1 * y1;
            bxy.x = x0 * y0; bxy.y = x1 * y1;
            acc1[0][g] = __builtin_amdgcn_wmma_f32_16x16x4_f32(false, bnd[kk], false, bx,  (short)0, acc1[0][g], false, false);
            acc1[1][g] = __builtin_amdgcn_wmma_f32_16x16x4_f32(false, bnd[kk], false, by,  (short)0, acc1[1][g], false, false);
            acc1[2][g] = __builtin_amdgcn_wmma_f32_16x16x4_f32(false, bnd[kk], false, bxx, (short)0, acc1[2][g], false, false);
            acc1[3][g] = __builtin_amdgcn_wmma_f32_16x16x4_f32(false, bnd[kk], false, byy, (short)0, acc1[3][g], false, false);
            acc1[4][g] = __builtin_amdgcn_wmma_f32_16x16x4_f32(false, bnd[kk], false, bxy, (short)0, acc1[4][g], false, false);
        }
    }

    // ---- Stage 2 per map: Out[:,16g..16g+15] = T1[:,16g..16g+23] * band chunks ----
    v8f outm[5][2];
    #pragma unroll
    for (int m = 0; m < 5; ++m) {
        __syncthreads();
        #pragma unroll
        for (int g = 0; g < 3; ++g)
            #pragma unroll
            for (int i = 0; i < 8; ++i)
                sT1[(i + 8 * hi) * T1W + 16 * g + lm] = acc1[m][g][i];
        __syncthreads();

        #pragma unroll
        for (int g = 0; g < 2; ++g) {
            v8f acc = {};
            #pragma unroll
            for (int kk = 0; kk < 6; ++kk) {
                const int cb = 16 * g + 4 * kk + 2 * hi;
                v2f a;
                a.x = sT1[lm * T1W + cb];
                a.y = sT1[lm * T1W + cb + 1];
                acc = __builtin_amdgcn_wmma_f32_16x16x4_f32(false, a, false, bnd[kk], (short)0, acc, false, false);
            }
            outm[m][g] = acc;
        }
    }

    // ---- SSIM per output element (D layout: row = i + 8*hi, col = 16*g + lm) ----
    const float drv   = DR[b];
    const float C1    = (0.01f * drv) * (0.01f * drv);
    const float C2    = (0.03f * drv) * (0.03f * drv);
    const float inv49 = 1.0f / 49.0f;
    const float covn  = 49.0f / 48.0f;

    float lsum = 0.0f;
    #pragma unroll
    for (int g = 0; g < 2; ++g) {
        const int col = c0 + 16 * g + lm;
        #pragma unroll
        for (int i = 0; i < 8; ++i) {
            const int row = r0 + i + 8 * hi;
            const float ux  = outm[0][g][i] * inv49;
            const float uy  = outm[1][g][i] * inv49;
            const float uxx = outm[2][g][i] * inv49;
            const float uyy = outm[3][g][i] * inv49;
            const float uxy = outm[4][g][i] * inv49;
            const float vx  = covn * (uxx - ux * ux);
            const float vy  = covn * (uyy - uy * uy);
            const float vxy = covn * (uxy - ux * uy);
            const float A1 = 2.0f * ux * uy + C1;
            const float A2 = 2.0f * vxy + C2;
            const float B1 = ux * ux + uy * uy + C1;
            const float B2 = vx + vy + C2;
            const float S  = (A1 * A2) * __builtin_amdgcn_rcpf(B1 * B2);
            if constexpr (EDGE) {
                if (row < OUT_H && col < OUT_W) lsum += S;
            } else {
                lsum += S;
            }
        }
    }

    // ---- wave32 reduction + one hashed atomic per tile ----
    #pragma unroll
    for (int off = 16; off > 0; off >>= 1)
        lsum += __shfl_xor(lsum, off, 32);
    if (lane == 0) {
        const unsigned cell = ((unsigned)tx + 20u * (unsigned)ty + 800u * (unsigned)b) & (NCELL - 1);
        atomicAdd(&ws[cell], lsum);
    }
}

// Interior tiles: x in [0,18], y in [0,38] -> no clamps, no masks.
__global__ __launch_bounds__(32)
void ssim_tile_interior(const float* __restrict__ X, const float* __restrict__ Y,
                        const float* __restrict__ DR, float* __restrict__ ws) {
    ssim_tile_body<false>(X, Y, DR, ws, blockIdx.x, blockIdx.y, blockIdx.z);
}

// Edge tiles: right strip x=19 (40 tiles) + bottom strip y=39, x in [0,18] (19 tiles).
__global__ __launch_bounds__(32)
void ssim_tile_edge(const float* __restrict__ X, const float* __restrict__ Y,
                    const float* __restrict__ DR, float* __restrict__ ws) {
    const int e = blockIdx.x;
    int tx, ty;
    if (e < TY) { tx = TX - 1; ty = e; }
    else        { tx = e - TY; ty = TY - 1; }
    ssim_tile_body<true>(X, Y, DR, ws, tx, ty, blockIdx.z);
}

extern "C" void kernel_launch(void* const* d_in, const int* in_sizes, int n_in,
                              void* d_out, int out_size, void* d_ws, size_t ws_size,
                              hipStream_t stream) {
    const float* X  = (const float*)d_in[0];   // (64,1,640,640) f32
    const float* Y  = (const float*)d_in[1];   // (64,1,640,640) f32
    const float* DR = (const float*)d_in[2];   // (64,) f32
    // d_in[3] = w (ones/49) — constant, folded into inv49.
    float* out = (float*)d_out;
    float* acc = (float*)d_ws;                 // NCELL float partial cells

    ssim_zero_kernel<<<NCELL / 256, 256, 0, stream>>>(acc);
    ssim_tile_interior<<<dim3(TX - 1, TY - 1, NB), 32, 0, stream>>>(X, Y, DR, acc);
    ssim_tile_edge<<<dim3(TY + TX - 1, 1, NB), 32, 0, stream>>>(X, Y, DR, acc);
    ssim_final_kernel<<<1, 256, 0, stream>>>(acc, out);
}